// VectorQuantizer_32289564131624
// MI455X (gfx1250) — compile-verified
//
#include <hip/hip_runtime.h>
#include <hip/hip_bf16.h>
#include <math.h>

// Problem constants (from reference): B=16, C=64, H=64, W=64, K=512
// N = B*H*W = 65536 rows, HW = 4096, total elements = 4194304.
#define NROWS   65536
#define CDIM    64
#define KCODES  512
#define HW      4096
#define TOTAL   4194304   // B*C*H*W

// LDS codebook staging: 128 codes per stage, stride 68 dwords (swizzle so a
// wave32 ds_load_b64 of a B fragment hits all 64 banks exactly once).
#define STAGE_K   128
#define LDS_STR   68      // dwords per code row (64 data + 4 pad)

typedef __attribute__((ext_vector_type(2))) float v2f;
typedef __attribute__((ext_vector_type(4))) float v4f;
typedef __attribute__((ext_vector_type(8))) float v8f;

// ---------------------------------------------------------------------------
// Kernel 1: per-code squared norms + zero the histogram / loss accumulator.
// grid = 2 blocks x 256 threads -> one thread per code.
// ---------------------------------------------------------------------------
__global__ void __launch_bounds__(256)
vq_init_enorm(const float* __restrict__ emb, float* __restrict__ enorm,
              unsigned* __restrict__ hist, float* __restrict__ loss)
{
    int k = blockIdx.x * 256 + threadIdx.x;   // 0..511
    const float* e = emb + k * CDIM;
    float s = 0.0f;
    #pragma unroll
    for (int c = 0; c < CDIM; ++c) { float v = e[c]; s += v * v; }
    enorm[k] = s;
    hist[k]  = 0u;
    if (k == 0) *loss = 0.0f;
}

// ---------------------------------------------------------------------------
// Kernel 2: WMMA distance GEMM + per-row argmin.
// One wave handles 32 consecutive rows (two 16-row WMMA tiles, same image b).
// The codebook is staged through LDS in 4 passes of 128 codes; B fragments are
// read with conflict-free ds_load_b64 and each feeds two chained
// V_WMMA_F32_16X16X4_F32 accumulations (one per row tile).
// Score = ||e_k||^2 - 2 * (x . e_k)   (row-constant ||x||^2 dropped).
// ---------------------------------------------------------------------------
__global__ void __launch_bounds__(256)
vq_argmin_kernel(const float* __restrict__ in, const float* __restrict__ emb,
                 const float* __restrict__ enorm, unsigned* __restrict__ hist,
                 float* __restrict__ idx_out)
{
    __shared__ float lds[STAGE_K * LDS_STR];   // 34,816 B

    const int tid     = threadIdx.x;
    const int lane    = tid & 31;
    const int wave    = tid >> 5;
    const int lm      = lane & 15;              // column / row-in-tile id
    const int hi      = lane >> 4;              // 0: lanes 0-15, 1: lanes 16-31
    const int rowBase = (blockIdx.x * 8 + wave) * 32;   // 32 rows per wave

    // flat[n][c] = in[((b*C + c)*HW) + hw], b = n>>12, hw = n & 4095.
    const int b  = rowBase >> 12;
    const int hw = (rowBase & (HW - 1)) + lm;
    const float* xb0 = in + (size_t)b * CDIM * HW + hw;        // row tile 0
    const float* xb1 = xb0 + 16;                               // row tile 1

    // A fragments (ISA f32 16x4 layout: lane<16 holds K=0,1; lane>=16 K=2,3).
    v2f a0[16], a1[16];
    #pragma unroll
    for (int kc = 0; kc < 16; ++kc) {
        const int c = 4 * kc + 2 * hi;
        a0[kc].x = xb0[(size_t)c * HW];
        a0[kc].y = xb0[(size_t)(c + 1) * HW];
        a1[kc].x = xb1[(size_t)c * HW];
        a1[kc].y = xb1[(size_t)(c + 1) * HW];
    }

    float bv0[8], bv1[8];
    int   bi0[8], bi1[8];
    #pragma unroll
    for (int i = 0; i < 8; ++i) {
        bv0[i] = 3.4e38f; bv1[i] = 3.4e38f; bi0[i] = 0; bi1[i] = 0;
    }

    for (int s = 0; s < KCODES / STAGE_K; ++s) {
        if (s) __syncthreads();   // previous stage fully consumed
        // ---- stage 128 codes into LDS (2 threads per code row) ----
        {
            const int r  = tid >> 1;             // 0..127
            const int c0 = (tid & 1) * 32;       // 0 or 32
            const float* g = emb + (size_t)(s * STAGE_K + r) * CDIM + c0;
            float* l = lds + r * LDS_STR + c0;
            #pragma unroll
            for (int j = 0; j < 8; ++j)
                *reinterpret_cast<v4f*>(l + 4 * j) =
                    *reinterpret_cast<const v4f*>(g + 4 * j);
        }
        __syncthreads();

        // ---- 8 column tiles of 16 codes each ----
        #pragma unroll
        for (int kt = 0; kt < STAGE_K / 16; ++kt) {
            const int kcol = s * STAGE_K + kt * 16 + lm;     // this lane's code
            const float* brow = lds + (kt * 16 + lm) * LDS_STR + 2 * hi;

            v8f acc0 = {0.f, 0.f, 0.f, 0.f, 0.f, 0.f, 0.f, 0.f};
            v8f acc1 = {0.f, 0.f, 0.f, 0.f, 0.f, 0.f, 0.f, 0.f};
            #pragma unroll
            for (int kc = 0; kc < 16; ++kc) {
                // B 4x16 fragment: lane<16 rows K=0,1 ; lane>=16 rows K=2,3.
                const v2f bfrag =
                    *reinterpret_cast<const v2f*>(brow + 4 * kc);
                acc0 = __builtin_amdgcn_wmma_f32_16x16x4_f32(
                           false, a0[kc], false, bfrag, (short)0, acc0,
                           false, false);
                acc1 = __builtin_amdgcn_wmma_f32_16x16x4_f32(
                           false, a1[kc], false, bfrag, (short)0, acc1,
                           false, false);
            }

            const float en = enorm[kcol];
            #pragma unroll
            for (int i = 0; i < 8; ++i) {
                const float s0 = fmaf(-2.0f, acc0[i], en);
                const float s1 = fmaf(-2.0f, acc1[i], en);
                if (s0 < bv0[i]) { bv0[i] = s0; bi0[i] = kcol; }
                if (s1 < bv1[i]) { bv1[i] = s1; bi1[i] = kcol; }
            }
        }
    }

    // Butterfly argmin across the 16 lanes holding a row's columns
    // (xor masks 1,2,4,8 stay inside each 16-lane half). Tie-break: lower idx
    // to match jnp.argmin first-occurrence semantics.
    #pragma unroll
    for (int m = 1; m <= 8; m <<= 1) {
        #pragma unroll
        for (int i = 0; i < 8; ++i) {
            float ov; int oi;
            ov = __shfl_xor(bv0[i], m, 32);
            oi = __shfl_xor(bi0[i], m, 32);
            if (ov < bv0[i] || (ov == bv0[i] && oi < bi0[i])) {
                bv0[i] = ov; bi0[i] = oi;
            }
            ov = __shfl_xor(bv1[i], m, 32);
            oi = __shfl_xor(bi1[i], m, 32);
            if (ov < bv1[i] || (ov == bv1[i] && oi < bi1[i])) {
                bv1[i] = ov; bi1[i] = oi;
            }
        }
    }

    // C/D layout: element i holds row M = i + 8*hi. Lane 0 emits rows 0-7 of
    // each tile, lane 16 emits rows 8-15.
    if (lm == 0) {
        #pragma unroll
        for (int i = 0; i < 8; ++i) {
            const int r0 = rowBase + 8 * hi + i;
            idx_out[r0] = (float)bi0[i];
            atomicAdd(&hist[bi0[i]], 1u);
            const int r1 = rowBase + 16 + 8 * hi + i;
            idx_out[r1] = (float)bi1[i];
            atomicAdd(&hist[bi1[i]], 1u);
        }
    }
}

// ---------------------------------------------------------------------------
// Kernel 3: gather emb[idx] back to BCHW + commitment-loss partial sums.
// out and in share the same linear BCHW index -> fully coalesced.
// ---------------------------------------------------------------------------
__global__ void __launch_bounds__(256)
vq_gather_loss(const float* __restrict__ in, const float* __restrict__ emb,
               const float* __restrict__ idxf, float* __restrict__ out,
               float* __restrict__ loss)
{
    const int o  = blockIdx.x * 256 + threadIdx.x;   // 0..TOTAL-1
    const int hw = o & (HW - 1);
    const int bc = o >> 12;
    const int c  = bc & (CDIM - 1);
    const int b  = bc >> 6;
    const int n  = (b << 12) | hw;

    const int   k = (int)idxf[n];
    const float q = emb[k * CDIM + c];
    const float x = in[o];
    out[o] = q;

    float d = q - x;
    d = d * d;
    #pragma unroll
    for (int m = 16; m >= 1; m >>= 1) d += __shfl_xor(d, m, 32);
    if ((threadIdx.x & 31) == 0) atomicAdd(loss, d);
}

// ---------------------------------------------------------------------------
// Kernel 4: finalize scalars: commitment_loss, perplexity, active_codes.
// ---------------------------------------------------------------------------
__global__ void __launch_bounds__(512)
vq_finalize(const unsigned* __restrict__ hist, const float* __restrict__ loss,
            const float* __restrict__ weight, float* __restrict__ scal)
{
    __shared__ float red[512];
    __shared__ float red2[512];
    const int k = threadIdx.x;   // 0..511
    const float p = (float)hist[k] * (1.0f / (float)NROWS);
    red[k]  = -p * logf(p + 1e-10f);
    red2[k] = (weight[k] >= 0.01f) ? 1.0f : 0.0f;
    __syncthreads();
    for (int s = 256; s > 0; s >>= 1) {
        if (k < s) { red[k] += red[k + s]; red2[k] += red2[k + s]; }
        __syncthreads();
    }
    if (k == 0) {
        scal[0] = *loss * (1.0f / (float)TOTAL);   // commitment_loss
        scal[1] = expf(red[0]);                    // perplexity
        scal[2] = red2[0];                         // active_codes
    }
}

// ---------------------------------------------------------------------------
// d_out layout (floats): [0, 4194304) quantized BCHW; [4194304] loss;
// [4194305] perplexity; [4194306] active_codes; [4194307, +65536) indices.
// d_ws layout (32-bit words): [0,512) enorm f32; [512,1024) hist u32;
// [1024] loss accumulator f32.
// ---------------------------------------------------------------------------
extern "C" void kernel_launch(void* const* d_in, const int* in_sizes, int n_in,
                              void* d_out, int out_size, void* d_ws, size_t ws_size,
                              hipStream_t stream)
{
    (void)in_sizes; (void)n_in; (void)out_size; (void)ws_size;

    const float* in  = (const float*)d_in[0];   // [16,64,64,64] f32
    const float* emb = (const float*)d_in[1];   // [512,64] f32
    const float* wgt = (const float*)d_in[2];   // [512] f32

    float*    O     = (float*)d_out;
    float*    enorm = (float*)d_ws;
    unsigned* hist  = (unsigned*)d_ws + 512;
    float*    loss  = (float*)d_ws + 1024;

    float* out  = O;
    float* scal = O + TOTAL;
    float* idxf = O + TOTAL + 3;

    vq_init_enorm   <<<2,            256, 0, stream>>>(emb, enorm, hist, loss);
    vq_argmin_kernel<<<NROWS / 256,  256, 0, stream>>>(in, emb, enorm, hist, idxf);
    vq_gather_loss  <<<TOTAL / 256,  256, 0, stream>>>(in, emb, idxf, out, loss);
    vq_finalize     <<<1,            512, 0, stream>>>(hist, loss, wgt, scal);
}